// PassiveLayer_85066122265462
// MI455X (gfx1250) — compile-verified
//
#include <hip/hip_runtime.h>
#include <stdint.h>

#define NXV 100
#define NYV 100
#define TABLE_N (NXV * NYV)
#define LW_F 10.0f
#define SIZE_F 0.1f
#define Z_TOP_F 1.0f
#define Z_BOT_F 0.9f
#define STEP_F 0.01f
#define N_STEPS 32
#define COEF_A_F 13.6e-3f
#define COEF_B_F 0.038f

__device__ __forceinline__ uint32_t rotl32(uint32_t v, int s) {
    return (v << s) | (v >> (32 - s));
}

// JAX threefry2x32: key schedule ks={k0,k1,k0^k1^0x1BD11BDA}, 20 rounds.
__device__ __forceinline__ void tf2x32(uint32_t k0, uint32_t k1,
                                       uint32_t c0, uint32_t c1,
                                       uint32_t& o0, uint32_t& o1) {
    const uint32_t ks2 = k0 ^ k1 ^ 0x1BD11BDAu;
    uint32_t x0 = c0 + k0, x1 = c1 + k1;
#define TFR(r) { x0 += x1; x1 = rotl32(x1, r); x1 ^= x0; }
    TFR(13) TFR(15) TFR(26) TFR(6)  x0 += k1;  x1 += ks2 + 1u;
    TFR(17) TFR(29) TFR(16) TFR(24) x0 += ks2; x1 += k0 + 2u;
    TFR(13) TFR(15) TFR(26) TFR(6)  x0 += k0;  x1 += k1 + 3u;
    TFR(17) TFR(29) TFR(16) TFR(24) x0 += k1;  x1 += ks2 + 4u;
    TFR(13) TFR(15) TFR(26) TFR(6)  x0 += ks2; x1 += k0 + 5u;
#undef TFR
    o0 = x0; o1 = x1;
}

// XLA's f32 ErfInv polynomial (Giles).
__device__ __forceinline__ float erfinv32(float x) {
    float w = -logf((1.0f - x) * (1.0f + x));
    float p;
    if (w < 5.0f) {
        w = w - 2.5f;
        p = 2.81022636e-08f;
        p = fmaf(p, w, 3.43273939e-07f);
        p = fmaf(p, w, -3.5233877e-06f);
        p = fmaf(p, w, -4.39150654e-06f);
        p = fmaf(p, w, 0.00021858087f);
        p = fmaf(p, w, -0.00125372503f);
        p = fmaf(p, w, -0.00417768164f);
        p = fmaf(p, w, 0.246640727f);
        p = fmaf(p, w, 1.50140941f);
    } else {
        w = sqrtf(w) - 3.0f;
        p = -0.000200214257f;
        p = fmaf(p, w, 0.000100950558f);
        p = fmaf(p, w, 0.00134934322f);
        p = fmaf(p, w, -0.00367342844f);
        p = fmaf(p, w, 0.00573950773f);
        p = fmaf(p, w, -0.0076224613f);
        p = fmaf(p, w, 0.00943887047f);
        p = fmaf(p, w, 1.00167406f);
        p = fmaf(p, w, 2.83297682f);
    }
    return p * x;
}

// raw bits -> uniform(nextafter(-1,0), 1) -> sqrt(2)*erfinv  (JAX normal)
__device__ __forceinline__ float bits_to_norm(uint32_t b) {
    float f = __uint_as_float((b >> 9) | 0x3f800000u) - 1.0f;  // [0,1)
    const float lo = -0.99999994f;                             // nextafter(-1,0)
    float u = fmaxf(fmaf(f, 2.0f, lo), lo);
    return 1.41421356f * erfinv32(u);
}

__global__ void __launch_bounds__(256)
muon_layer_kernel(const float* __restrict__ gx,  const float* __restrict__ gy,
                  const float* __restrict__ gz,  const float* __restrict__ gtx,
                  const float* __restrict__ gty, const float* __restrict__ gmom,
                  const float* __restrict__ grad, float4* __restrict__ out,
                  int ntot) {
    __shared__ float s_rad[TABLE_N];
    __shared__ uint2 s_keys[N_STEPS];

    // Per-block: fold_in(key(42), i) for each step, by the first wave.
    if (threadIdx.x < N_STEPS) {
        uint32_t o0, o1;
        tf2x32(0u, 42u, 0u, (uint32_t)threadIdx.x, o0, o1);
        s_keys[threadIdx.x] = make_uint2(o0, o1);
    }

    // Fill the 40KB voxel table via CDNA5 async global->LDS DMA (ASYNCcnt).
    // LDS byte address = low 32 bits of the generic pointer (ISA aperture rule).
    for (int idx = threadIdx.x; idx < TABLE_N; idx += blockDim.x) {
        uint32_t lds_off = (uint32_t)(uintptr_t)(&s_rad[idx]);
        uint64_t gaddr   = (uint64_t)(uintptr_t)(grad + idx);
        asm volatile("global_load_async_to_lds_b32 %0, %1, off"
                     :: "v"(lds_off), "v"(gaddr)
                     : "memory");
    }
    asm volatile("s_wait_asynccnt 0" ::: "memory");
    __syncthreads();

    const int n = blockIdx.x * blockDim.x + threadIdx.x;
    if (n >= ntot) return;

    float px = gx[n], py = gy[n], pz = gz[n];
    float tx = gtx[n], ty = gty[n];
    const float p = gmom[n];

    // propagate to top of layer
    float dz0 = pz - Z_TOP_F;
    px += dz0 * tanf(tx);
    py += dz0 * tanf(ty);
    pz -= dz0;

    const uint32_t un = (uint32_t)n;
    const uint32_t NT = (uint32_t)ntot;
    const float coef_over_p = COEF_A_F / p;

#pragma unroll 1
    for (int i = 0; i < N_STEPS; ++i) {
        const bool mask = (pz > Z_BOT_F) && (pz <= Z_TOP_F);

        // Once every lane of this wave has left the layer, no state can change
        // for the remaining steps (all updates are mask-gated and the exit snap
        // lands pz exactly on Z_BOT) -> bit-exact early out. Saves ~half of the
        // Threefry/erfinv/tan work since muons exit after ~10-14 of 32 steps.
        if (__builtin_amdgcn_ballot_w32(mask) == 0u) break;

        const uint2 k = s_keys[i];

        const float ttx = tanf(tx);
        const float tty = tanf(ty);
        const float cth = rsqrtf(1.0f + ttx * ttx + tty * tty);
        const bool smask = (px >= 0.0f) && (px < LW_F) &&
                           (py >= 0.0f) && (py < LW_F) && mask;

        const int ix = (int)fminf(fmaxf(floorf(px / SIZE_F), 0.0f), (float)(NXV - 1));
        const int iy = (int)fminf(fmaxf(floorf(py / SIZE_F), 0.0f), (float)(NYV - 1));
        const float x0v = s_rad[ix * NYV + iy];   // ds_load_b32 gather

        const float r_out = (pz - Z_BOT_F) / cth;
        const float sl = fmaxf(fminf(STEP_F, r_out), 1e-9f);
        const float nx0 = sl / x0v;
        const float th0 = coef_over_p * sqrtf(nx0) * fmaf(COEF_B_F, logf(nx0), 1.0f);

        // zr layout for shape (4,N): threefry counters (n, n+2N) -> (zr0,zr2),
        // (n+N, n+3N) -> (zr1,zr3)
        uint32_t b0, b1, b2, b3;
        tf2x32(k.x, k.y, un,      un + 2u * NT, b0, b2);
        tf2x32(k.x, k.y, un + NT, un + 3u * NT, b1, b3);
        const float zr0 = bits_to_norm(b0);
        const float zr1 = bits_to_norm(b1);
        const float zr2 = bits_to_norm(b2);
        const float zr3 = bits_to_norm(b3);

        const float dtx = th0 * zr0;
        const float dty = th0 * zr1;
        const float dxv = sl * th0 * (zr2 / 3.4641016f + zr0 * 0.5f);
        const float dyv = sl * th0 * (zr3 / 3.4641016f + zr1 * 0.5f);

        if (smask) { px += dxv; py += dyv; }

        const float dz = STEP_F * cth;
        if (mask) { px += dz * ttx; py += dz * tty; pz -= dz; }

        const float dzr = pz - Z_BOT_F;
        if ((dzr < 0.0f) && mask) { px += dzr * ttx; py += dzr * tty; pz -= dzr; }

        if (smask) { tx += dtx; ty += dty; }
    }

    // snap to bottom of layer
    const float dz1 = pz - Z_BOT_F;
    px += dz1 * tanf(tx);
    py += dz1 * tanf(ty);

    out[n] = make_float4(px, py, tx, ty);
}

extern "C" void kernel_launch(void* const* d_in, const int* in_sizes, int n_in,
                              void* d_out, int out_size, void* d_ws, size_t ws_size,
                              hipStream_t stream) {
    const float* x    = (const float*)d_in[0];
    const float* y    = (const float*)d_in[1];
    const float* z    = (const float*)d_in[2];
    const float* thx  = (const float*)d_in[3];
    const float* thy  = (const float*)d_in[4];
    const float* mom  = (const float*)d_in[5];
    const float* rad  = (const float*)d_in[6];
    float4* out = (float4*)d_out;

    const int n = in_sizes[0];
    const int block = 256;
    const int grid = (n + block - 1) / block;
    muon_layer_kernel<<<grid, block, 0, stream>>>(x, y, z, thx, thy, mom, rad, out, n);
}